// SAGE_three_layers_23235773072077
// MI455X (gfx1250) — compile-verified
//
#include <hip/hip_runtime.h>

// GraphSAGE 3-layer forward for MI455X (gfx1250, wave32).
// Aggregation is the bandwidth roofline (~5 GB irregular traffic @ 23.3 TB/s);
// GEMMs (~20 GFLOP) run on V_WMMA_F32_16X16X4_F32 to keep f32 precision.
// Weights staged K-major in LDS so each B fragment is one ds_load_b64 into an
// even-aligned VGPR pair (no shuffle movs), conflict-free with stride KCH+4.

typedef __attribute__((ext_vector_type(2))) float v2f;
typedef __attribute__((ext_vector_type(8))) float v8f;

#define DI 128  // input feature dim for every layer's aggregation / GEMM K-half

// ---------------------------------------------------------------- utilities
__global__ void zero_f32_kernel(float* __restrict__ p, int n) {
  int i = (blockIdx.x * blockDim.x + threadIdx.x) * 4;
  if (i + 3 < n) {
    float4 z; z.x = 0.f; z.y = 0.f; z.z = 0.f; z.w = 0.f;
    *(float4*)(p + i) = z;
  } else {
    for (int k = i; k < n; ++k) p[k] = 0.f;
  }
}

__global__ void deg_kernel(const int* __restrict__ dst, float* __restrict__ deg, int E) {
  int e = blockIdx.x * blockDim.x + threadIdx.x;
  if (e < E) atomicAdd(&deg[dst[e]], 1.0f);
}

__global__ void invdeg_kernel(float* __restrict__ deg, int N) {
  int i = blockIdx.x * blockDim.x + threadIdx.x;
  if (i < N) deg[i] = 1.0f / fmaxf(deg[i], 1.0f);
}

// One wave per edge: 32 lanes x float4 = 128 features. Coalesced gather of the
// source row, contiguous f32 atomic scatter-add into the destination row.
__global__ void scatter_kernel(const float* __restrict__ h, const int* __restrict__ src,
                               const int* __restrict__ dst, float* __restrict__ agg, int E) {
  long tid = (long)blockIdx.x * blockDim.x + threadIdx.x;
  if (tid >= (long)E * (DI / 4)) return;
  int e = (int)(tid >> 5);          // DI/4 == 32 chunks per edge
  int c = ((int)tid & 31) * 4;
  int s = src[e], d = dst[e];
  const float4 v = *(const float4*)(h + (long)s * DI + c);
  float* p = agg + (long)d * DI + c;
  atomicAdd(p + 0, v.x);
  atomicAdd(p + 1, v.y);
  atomicAdd(p + 2, v.z);
  atomicAdd(p + 3, v.w);
}

// ------------------------------------------------------------------- GEMM
// out[r,:] = h[r,:] @ Wself + (agg[r,:] * invdeg[r]) @ Wneigh + b
// optionally followed by LayerNorm + ReLU (fused).
// Block: 256 threads = 8 waves; wave w owns rows [blk*128 + w*16, +16).
// K = 2*DI = 256 (concat of self/neighbor halves), stepped 4 per WMMA.
template <int DO, bool LN>
__global__ __launch_bounds__(256) void sage_gemm_kernel(
    const float* __restrict__ h, const float* __restrict__ agg,
    const float* __restrict__ invdeg,
    const float* __restrict__ wself, const float* __restrict__ wneigh,
    const float* __restrict__ bias,
    const float* __restrict__ g, const float* __restrict__ beta,
    float* __restrict__ out, int N) {
  constexpr int JT   = DO / 16;   // 16-wide output column tiles per wave
  constexpr int KCH  = 64;        // K rows staged per LDS chunk
  constexpr int KSTR = KCH + 4;   // K-major stride (floats): bank = (4n+kc)%64,
                                  // all 32 b64 reads hit 64 distinct banks
  __shared__ float wlds[DO * KSTR];  // DO=128: 34 KB, DO=64: 17 KB

  const int tid   = threadIdx.x;
  const int wave  = tid >> 5;
  const int lane  = tid & 31;
  const int ncol  = lane & 15;    // column within a 16-wide tile
  const int khalf = lane >> 4;    // A/B fragments: lanes 16..31 take K+2
  const int row_base = blockIdx.x * 128 + wave * 16;
  const int arow = min(row_base + ncol, N - 1);  // clamped load row (stores guarded)

  const float* hrow = h   + (long)arow * DI;
  const float* nrow = agg + (long)arow * DI;
  const float  inv  = invdeg[arow];

  v8f acc[JT];
#pragma unroll
  for (int j = 0; j < JT; ++j)
#pragma unroll
    for (int i = 0; i < 8; ++i) acc[j][i] = 0.0f;

  for (int kbase = 0; kbase < 2 * DI; kbase += KCH) {
    __syncthreads();
    // stage KCH rows of [Wself; Wneigh], transposed to K-major in LDS:
    // wlds[n*KSTR + k] so B pairs (k, k+1) are contiguous (one ds_load_b64)
    for (int idx = tid; idx < KCH * DO; idx += 256) {
      const int kr = idx / DO;          // 0..KCH-1
      const int nn = idx - kr * DO;     // coalesced global read along nn
      const int kk = kbase + kr;
      const float wv = (kk < DI) ? wself[kk * DO + nn] : wneigh[(kk - DI) * DO + nn];
      wlds[nn * KSTR + kr] = wv;
    }
    __syncthreads();

    const bool  neigh = (kbase >= DI);                      // uniform per chunk
    const float* abase = neigh ? (nrow + (kbase - DI)) : (hrow + kbase);
    const float  asc   = neigh ? inv : 1.0f;                // fold mean-divide into A

#pragma unroll 4
    for (int kk = 0; kk < KCH; kk += 4) {
      const int kc = kk + 2 * khalf;
      // A frag (16x4 f32): lane = row, vgpr0/1 = K kc/kc+1 (halves offset by 2)
      v2f a = *(const v2f*)(abase + kc);
      a.x *= asc; a.y *= asc;
#pragma unroll
      for (int j = 0; j < JT; ++j) {
        // B frag (4x16 f32): one b64 LDS load -> (w[kc][n], w[kc+1][n])
        const v2f bb = *(const v2f*)(&wlds[(j * 16 + ncol) * KSTR + kc]);
        acc[j] = __builtin_amdgcn_wmma_f32_16x16x4_f32(
            false, a, false, bb, (short)0, acc[j], false, false);
      }
    }
  }

  // ---- epilogue: bias (+ LayerNorm + ReLU) ----
#pragma unroll
  for (int j = 0; j < JT; ++j) {
    const float bj = bias[j * 16 + ncol];
#pragma unroll
    for (int i = 0; i < 8; ++i) acc[j][i] += bj;
  }

  if constexpr (LN) {
    // C layout: vgpr i holds row i (lanes 0-15) / row i+8 (lanes 16-31);
    // a row's DO values live across 16 lanes x JT vgpr-slots -> shfl_xor masks
    // 1..8 reduce exactly within each 16-lane half.
    float mu[8], rs[8];
#pragma unroll
    for (int i = 0; i < 8; ++i) {
      float s = 0.f, q = 0.f;
#pragma unroll
      for (int j = 0; j < JT; ++j) { const float v = acc[j][i]; s += v; q += v * v; }
#pragma unroll
      for (int m = 8; m >= 1; m >>= 1) {
        s += __shfl_xor(s, m, 32);
        q += __shfl_xor(q, m, 32);
      }
      const float m1 = s * (1.0f / DO);
      mu[i] = m1;
      rs[i] = rsqrtf(q * (1.0f / DO) - m1 * m1 + 1e-5f);
    }
#pragma unroll
    for (int j = 0; j < JT; ++j) {
      const float gj  = g[j * 16 + ncol];
      const float btj = beta[j * 16 + ncol];
#pragma unroll
      for (int i = 0; i < 8; ++i) {
        const float v = (acc[j][i] - mu[i]) * rs[i] * gj + btj;
        acc[j][i] = fmaxf(v, 0.0f);
      }
    }
  }

#pragma unroll
  for (int i = 0; i < 8; ++i) {
    const int r = row_base + i + 8 * khalf;
    if (r < N) {
#pragma unroll
      for (int j = 0; j < JT; ++j)
        out[(long)r * DO + j * 16 + ncol] = acc[j][i];
    }
  }
}

// ---------------------------------------------------------------- launcher
static inline int cdiv_i(long a, long b) { return (int)((a + b - 1) / b); }

extern "C" void kernel_launch(void* const* d_in, const int* in_sizes, int n_in,
                              void* d_out, int out_size, void* d_ws, size_t ws_size,
                              hipStream_t stream) {
  const float* x    = (const float*)d_in[0];
  const int*   esrc = (const int*)d_in[1];
  const int*   edst = (const int*)d_in[2];
  const float* ws0  = (const float*)d_in[3];
  const float* wn0  = (const float*)d_in[4];
  const float* b0   = (const float*)d_in[5];
  const float* ws1  = (const float*)d_in[6];
  const float* wn1  = (const float*)d_in[7];
  const float* b1   = (const float*)d_in[8];
  const float* ws2  = (const float*)d_in[9];
  const float* wn2  = (const float*)d_in[10];
  const float* b2   = (const float*)d_in[11];
  const float* g0   = (const float*)d_in[12];
  const float* be0  = (const float*)d_in[13];
  const float* g1   = (const float*)d_in[14];
  const float* be1  = (const float*)d_in[15];
  float* out = (float*)d_out;

  const int N = in_sizes[0] / DI;
  const int E = in_sizes[1];
  (void)n_in; (void)out_size; (void)ws_size;

  // workspace carve: deg/inv (N), agg (N*DI), h1 (N*DI), h2 (N*DI)  ~154 MB
  char* base = (char*)d_ws;
  size_t off = 0;
  auto carve = [&](size_t bytes) -> float* {
    float* p = (float*)(base + off);
    off += (bytes + 255) & ~(size_t)255;
    return p;
  };
  float* deg = carve((size_t)N * sizeof(float));
  float* agg = carve((size_t)N * DI * sizeof(float));
  float* h1  = carve((size_t)N * DI * sizeof(float));
  float* h2  = carve((size_t)N * DI * sizeof(float));

  const int T = 256;
  const int nFeat   = N * DI;
  const int gZeroF  = cdiv_i(nFeat, (long)T * 4);
  const int gZeroN  = cdiv_i(N, (long)T * 4);
  const int gScat   = cdiv_i((long)E * (DI / 4), T);
  const int gGemm   = cdiv_i(N, 128);

  // degree -> 1/max(deg,1), once
  zero_f32_kernel<<<gZeroN, T, 0, stream>>>(deg, N);
  deg_kernel<<<cdiv_i(E, T), T, 0, stream>>>(edst, deg, E);
  invdeg_kernel<<<cdiv_i(N, T), T, 0, stream>>>(deg, N);

  // layer 0: x -> h1 (LN+ReLU)
  zero_f32_kernel<<<gZeroF, T, 0, stream>>>(agg, nFeat);
  scatter_kernel<<<gScat, T, 0, stream>>>(x, esrc, edst, agg, E);
  sage_gemm_kernel<128, true><<<gGemm, T, 0, stream>>>(
      x, agg, deg, ws0, wn0, b0, g0, be0, h1, N);

  // layer 1: h1 -> h2 (LN+ReLU)
  zero_f32_kernel<<<gZeroF, T, 0, stream>>>(agg, nFeat);
  scatter_kernel<<<gScat, T, 0, stream>>>(h1, esrc, edst, agg, E);
  sage_gemm_kernel<128, true><<<gGemm, T, 0, stream>>>(
      h1, agg, deg, ws1, wn1, b1, g1, be1, h2, N);

  // layer 2: h2 -> out (no norm/act), DO=64
  zero_f32_kernel<<<gZeroF, T, 0, stream>>>(agg, nFeat);
  scatter_kernel<<<gScat, T, 0, stream>>>(h2, esrc, edst, agg, E);
  sage_gemm_kernel<64, false><<<gGemm, T, 0, stream>>>(
      h2, agg, deg, ws2, wn2, b2, nullptr, nullptr, out, N);
}